// EquivariantUpsampleConv_51926154609192
// MI455X (gfx1250) — compile-verified
//
#include <hip/hip_runtime.h>
#include <hip/hip_bf16.h>

typedef float v2f __attribute__((ext_vector_type(2)));
typedef float v8f __attribute__((ext_vector_type(8)));

// D = A(16x4,f32) * B(4x16,f32) + C(16x16,f32), full f32 precision (matches reference)
__device__ __forceinline__ v8f wmma_f32(v2f a, v2f b, v8f c) {
    return __builtin_amdgcn_wmma_f32_16x16x4_f32(
        /*neg_a=*/false, a, /*neg_b=*/false, b,
        /*c_mod=*/(short)0, c, /*reuse_a=*/false, /*reuse_b=*/false);
}

// ---------------------------------------------------------------------------
// Prep: fold learned path weights + SH kernel into two padded GEMM operands.
//   Wtp_p[24][496] : Wtp_p[i][j*22+k] = sum_p w_tp[p]*cg_tp[p,i,j,k]   (zero-padded)
//   Kp[88][32]     : Kp[j*22+i][k]    = sum_s sh[j,s] * sum_p w_agg[p]*cg_agg[p,i,s,k]
// ---------------------------------------------------------------------------
__global__ void equp_prep_kernel(const float* __restrict__ sh,
                                 const float* __restrict__ cg_agg,
                                 const float* __restrict__ cg_tp,
                                 const float* __restrict__ w_agg,
                                 const float* __restrict__ w_tp,
                                 float* __restrict__ Wtp_p,
                                 float* __restrict__ Kp) {
    const int NW = 24 * 496;
    const int NK = 88 * 32;
    for (int e = blockIdx.x * blockDim.x + threadIdx.x; e < NW + NK;
         e += gridDim.x * blockDim.x) {
        if (e < NW) {
            int i = e / 496, col = e % 496;
            float v = 0.f;
            if (i < 22 && col < 484) {
                int j = col / 22, kk = col % 22;
#pragma unroll
                for (int p = 0; p < 8; ++p)
                    v += w_tp[p] * cg_tp[((p * 22 + i) * 22 + j) * 22 + kk];
            }
            Wtp_p[e] = v;
        } else {
            int e2 = e - NW;
            int row = e2 / 32, k = e2 % 32;
            float v = 0.f;
            if (k < 22) {
                int j = row / 22, i = row % 22;
#pragma unroll
                for (int s = 0; s < 6; ++s) {
                    float wag = 0.f;
#pragma unroll
                    for (int p = 0; p < 6; ++p)
                        wag += w_agg[p] * cg_agg[((p * 22 + i) * 6 + s) * 22 + k];
                    v += sh[j * 6 + s] * wag;
                }
            }
            Kp[e2] = v;
        }
    }
}

// ---------------------------------------------------------------------------
// Main kernel: one 256-thread (8-wave) WG per 16-lr-pixel row segment.
//   Phase A : A(16,496) = F(16,24) @ Wtp_p(24,496)         -> sA[blk][j][k]
//   Phase U : U(48,128) = Fstk(48,24) @ [K0|K1|K2|K3]      -> sU[pix][j][k]
//             (rows 0..16 = lr row by, 17..33 = lr row by+1, rest zero)
//   Phase B : per block: 4 context classes from sU, then
//             O(16,32) = C(16,24) @ sA[blk](24,32) + residual
// ---------------------------------------------------------------------------
#define SEG 16

__global__ __launch_bounds__(256)
void equp_main_kernel(const float* __restrict__ f4, const float* __restrict__ f6,
                      const float* __restrict__ Wtp_p, const float* __restrict__ Kp,
                      float* __restrict__ out4, float* __restrict__ out6,
                      int H, int W) {
    __shared__ float sFst[48][24];        // stacked lr feats (row0: 0..16, row1: 17..33)
    __shared__ float sKcat[24][128];      // [K0|K1|K2|K3] B-operand, sKcat[i][j*32+k]
    __shared__ float sA[SEG][24][32];     // per-block A_b (final GEMM B-operand)
    __shared__ float sU[48][4][32];       // u_j = f_pix @ K_j (rows >= 34 are zero pad)
    __shared__ float sCt[8][4][24];       // per-wave context classes (A-operand rows)

    const int tid  = threadIdx.x;
    const int lane = tid & 31;
    const int wave = tid >> 5;
    const int m    = lane & 15;   // A-operand row / B,D-operand column
    const int hi   = lane >> 4;   // half-wave select
    const int kof  = hi * 2;      // K offset per ISA A/B layout (hi half holds K+2)

    const int segs = W >> 4;
    const int by   = blockIdx.x / segs;
    const int bx0  = (blockIdx.x % segs) << 4;
    const int Wr   = W * 4;

    // ---- stage stacked feature rows (replicate-pad clamped), pad to 24 ----
    for (int e = tid; e < 48 * 24; e += 256) {
        int row = e / 24, i = e % 24;
        float v = 0.f;
        if (row < 34 && i < 22) {
            int which = row >= 17;
            int t = which ? row - 17 : row;
            int bx = bx0 + t; if (bx > W - 1) bx = W - 1;
            int byy = which ? ((by + 1 > H - 1) ? H - 1 : by + 1) : by;
            long b = (long)byy * W + bx;
            v = (i < 9) ? f4[b * 9 + i] : f6[b * 13 + (i - 9)];
        }
        (&sFst[0][0])[e] = v;
    }
    // ---- stage concatenated context operand [K0|K1|K2|K3] ----
    for (int e = tid; e < 24 * 128; e += 256) {
        int i = e / 128, c = e % 128;
        int j = c >> 5, k = c & 31;
        (&sKcat[0][0])[e] = (i < 22) ? Kp[(j * 22 + i) * 32 + k] : 0.f;
    }
    for (int e = tid; e < SEG * 24 * 32; e += 256) (&sA[0][0][0])[e] = 0.f;
    // prefetch the streamed B-operand of Phase A (lowers to global_prefetch_b8)
    __builtin_prefetch(Wtp_p + tid * 46, 0, 0);
    __syncthreads();

    // ---- Phases A + U: 31 A-tiles then 24 U-tiles, round-robin over waves ----
    for (int task = wave; task < 31 + 24; task += 8) {
        v8f acc = {};
        if (task < 31) {                       // A-GEMM n-tile
            const int nt = task;
#pragma unroll
            for (int ks = 0; ks < 6; ++ks) {
                v2f a, b;
                a.x = sFst[m][ks * 4 + kof];
                a.y = sFst[m][ks * 4 + kof + 1];
                const float* bp = Wtp_p + nt * 16 + m;
                b.x = bp[(ks * 4 + kof) * 496];
                b.y = bp[(ks * 4 + kof + 1) * 496];
                acc = wmma_f32(a, b, acc);
            }
            // cols 484..495 are exact zeros and land in sA[b][22][0..11]
            // (a required-zero pad region), so the scatter is unconditional.
#pragma unroll
            for (int r = 0; r < 8; ++r) {
                int brow = r + hi * 8;         // lr pixel within segment
                int col  = nt * 16 + m;
                int j = col / 22, kk = col % 22;
                sA[brow][j][kk] = acc[r];      // scatter into [j][k] form
            }
        } else {                               // U-GEMM tile
            const int id = task - 31;
            const int mt = id >> 3, nt = id & 7;
#pragma unroll
            for (int ks = 0; ks < 6; ++ks) {
                v2f a, b;
                a.x = sFst[mt * 16 + m][ks * 4 + kof];
                a.y = sFst[mt * 16 + m][ks * 4 + kof + 1];
                b.x = sKcat[ks * 4 + kof][nt * 16 + m];
                b.y = sKcat[ks * 4 + kof + 1][nt * 16 + m];
                acc = wmma_f32(a, b, acc);
            }
            // rows 34..47 are zeros into the sU pad rows: store unconditionally.
#pragma unroll
            for (int r = 0; r < 8; ++r) {
                int mrow = mt * 16 + r + hi * 8;
                int ncol = nt * 16 + m;
                sU[mrow][ncol >> 5][ncol & 31] = acc[r];
            }
        }
    }
    __syncthreads();

    // ---- Phase B: per block, assemble 4 context classes + final GEMM ----
    const int sy = m >> 2, sx = m & 3;                 // hr sub-position of row m
    const int cls_m = (((sy == 3) ? 2 : 0) | ((sx == 3) ? 1 : 0));
    for (int blk = wave * 2; blk < wave * 2 + 2; ++blk) {
        // context class c[cls][kk] = u0[b] + u1[b+cs] + u2[rs+b] + u3[rs+b+cs]
        for (int e = lane; e < 4 * 24; e += 32) {
            int cls = e / 24, kk = e % 24;
            int rs = (cls & 2) ? 17 : 0;
            int cs = (cls & 1) ? 1 : 0;
            sCt[wave][cls][kk] = sU[blk][0][kk] + sU[blk + cs][1][kk] +
                                 sU[rs + blk][2][kk] + sU[rs + blk + cs][3][kk];
        }
        asm volatile("s_wait_dscnt 0" ::: "memory");   // wave-private LDS RAW
        // final: O(16,32) = C(16,24) @ A_b(24,32)
        v8f o0 = {}, o1 = {};
#pragma unroll
        for (int ks = 0; ks < 6; ++ks) {
            v2f a, b0, b1;
            a.x = sCt[wave][cls_m][ks * 4 + kof];
            a.y = sCt[wave][cls_m][ks * 4 + kof + 1];
            b0.x = sA[blk][ks * 4 + kof][m];       b0.y = sA[blk][ks * 4 + kof + 1][m];
            b1.x = sA[blk][ks * 4 + kof][16 + m];  b1.y = sA[blk][ks * 4 + kof + 1][16 + m];
            o0 = wmma_f32(a, b0, o0);
            o1 = wmma_f32(a, b1, o1);
        }
        // residual add + scatter into the two output arrays (out[:, :9], out[:, 9:])
#pragma unroll
        for (int r = 0; r < 8; ++r) {
            int mm = r + hi * 8;
            int syy = mm >> 2, sxx = mm & 3;
            long y = (long)by * 4 + syy;
            long x = (long)(bx0 + blk) * 4 + sxx;
            long n = y * Wr + x;
            int k0 = m;
            float v0 = o0[r] + sFst[blk][k0];
            if (k0 < 9) out4[n * 9 + k0] = v0;
            else        out6[n * 13 + (k0 - 9)] = v0;
            int k1 = 16 + m;
            if (k1 < 22) {
                float v1 = o1[r] + sFst[blk][k1];
                out6[n * 13 + (k1 - 9)] = v1;
            }
        }
    }
}

extern "C" void kernel_launch(void* const* d_in, const int* in_sizes, int n_in,
                              void* d_out, int out_size, void* d_ws, size_t ws_size,
                              hipStream_t stream) {
    const float* f4     = (const float*)d_in[0];
    const float* f6     = (const float*)d_in[1];
    const float* sh     = (const float*)d_in[2];
    const float* cg_agg = (const float*)d_in[3];
    const float* cg_tp  = (const float*)d_in[4];
    const float* w_agg  = (const float*)d_in[5];
    const float* w_tp   = (const float*)d_in[6];

    int HW = in_sizes[0] / 9;
    int H = 1;
    while ((long)(H + 1) * (H + 1) <= (long)HW) ++H;  // isqrt: H == W in reference
    int W = HW / H;

    float* Wtp_p = (float*)d_ws;
    float* Kp    = Wtp_p + 24 * 496;
    float* out4  = (float*)d_out;
    float* out6  = out4 + (long)HW * 16 * 9;

    equp_prep_kernel<<<8, 256, 0, stream>>>(sh, cg_agg, cg_tp, w_agg, w_tp, Wtp_p, Kp);
    int blocks = H * (W / 16);
    equp_main_kernel<<<blocks, 256, 0, stream>>>(f4, f6, Wtp_p, Kp, out4, out6, H, W);
}